// MultiHaedAttention_24197845746171
// MI455X (gfx1250) — compile-verified
//
#include <hip/hip_runtime.h>
#include <math.h>

typedef float v2f __attribute__((ext_vector_type(2)));
typedef float v4f __attribute__((ext_vector_type(4)));
typedef float v8f __attribute__((ext_vector_type(8)));
typedef int   v4i __attribute__((ext_vector_type(4)));

// D = A(16x4) * B(4x16) + C(16x16), all f32, wave32 CDNA5 WMMA
#define WMMA4F32(a, b, c) \
  __builtin_amdgcn_wmma_f32_16x16x4_f32(false, (a), false, (b), (short)0, (c), false, false)

// Probe for the gfx1250 async global->LDS copy path (ASYNCcnt-tracked).
#if defined(__has_builtin)
#if __has_builtin(__builtin_amdgcn_global_load_async_to_lds_b128) && \
    __has_builtin(__builtin_amdgcn_s_wait_asynccnt)
#define HAVE_ASYNC_LDS 1
#endif
#endif
#ifndef HAVE_ASYNC_LDS
#define HAVE_ASYNC_LDS 0
#endif

constexpr int Bc  = 8;       // batch
constexpr int Nn  = 2048;    // sequence length
constexpr int D   = 64;      // per-head dim
constexpr int H   = 8;       // heads
constexpr int HD  = H * D;   // 512
constexpr int QKV = 3 * HD;  // 1536

constexpr int KSTR = 68;     // LDS K-tile row stride (dwords): banks 4m+k0, conflict-free
constexpr int VSTR = 72;     // LDS V-tile row stride (dwords): banks 8kp+c+m, conflict-free

#if HAVE_ASYNC_LDS
typedef __attribute__((address_space(1))) v4i g_v4i;  // global int4
typedef __attribute__((address_space(3))) v4i l_v4i;  // LDS int4

__device__ __forceinline__ void async_g2l_b128(const float* g, float* l) {
  __builtin_amdgcn_global_load_async_to_lds_b128(
      (g_v4i*)g, (l_v4i*)l, /*offset=*/0, /*cpol=*/0);
}
#endif

// ---------------------------------------------------------------------------
// Stage 1: qkv = x @ W_qkv   ->  Q/K/V in [b, h, n, d] layout
// One 16x16 output tile per wave; K = 64 -> 16 wmma_f32_16x16x4 steps.
// ---------------------------------------------------------------------------
__global__ __launch_bounds__(128) void mha_qkv_proj(
    const float* __restrict__ x,      // [B*N, 64]
    const float* __restrict__ Wqkv,   // [64, 1536]
    float* __restrict__ Q, float* __restrict__ K, float* __restrict__ V) {
  const int lane = threadIdx.x & 31;
  const int wave = threadIdx.x >> 5;
  const int m    = lane & 15;
  const int lh   = lane >> 4;

  const int nTilesC = QKV / 16;                 // 96
  const int tile    = blockIdx.x * 4 + wave;    // 1024*96 tiles total
  const int mt      = tile / nTilesC;
  const int ct      = tile % nTilesC;

  const float* xrow = x + (size_t)(mt * 16 + m) * D;  // A-fragment row for this lane
  const int c = ct * 16 + m;                          // B-fragment column for this lane

  v8f acc = {0.f, 0.f, 0.f, 0.f, 0.f, 0.f, 0.f, 0.f};
#pragma unroll
  for (int kk = 0; kk < D; kk += 4) {
    v2f a = *(const v2f*)(xrow + kk + 2 * lh);
    v2f b;
    b.x = Wqkv[(size_t)(kk + 2 * lh + 0) * QKV + c];
    b.y = Wqkv[(size_t)(kk + 2 * lh + 1) * QKV + c];
    acc = WMMA4F32(a, b, acc);
  }

  // Scatter C tile (row = i + 8*lh, col = lane&15) to head-major [b,h,n,d].
  const int cbase = ct * 16;
  const int sect  = cbase / HD;                 // 0=Q 1=K 2=V (tile never straddles)
  const int hh    = (cbase % HD) / D;
  const int d0    = (cbase % D) + m;            // per-lane d within head
  float* dst = (sect == 0) ? Q : ((sect == 1) ? K : V);
  const int gb    = (mt * 16) / Nn;             // batch (tile never straddles)
  const int nbase = mt * 16 - gb * Nn;
#pragma unroll
  for (int i = 0; i < 8; ++i) {
    const int n = nbase + i + 8 * lh;
    dst[(size_t)((gb * H + hh) * Nn + n) * D + d0] = acc[i];
  }
}

// ---------------------------------------------------------------------------
// Stage 2: flash attention. 8 waves/WG (256 thr) own 128 query rows of one
// (b,h). K/V tiles (16 keys) staged in double-buffered LDS shared by all 8
// waves (8x less WGP-ingress traffic) — via GLOBAL_LOAD_ASYNC_TO_LDS_B128
// when the toolchain exposes it (no VGPR round-trip, ASYNCcnt-overlapped),
// else via coalesced b128 register bounce. Online softmax per 16-row Q tile.
// Output z written IN PLACE over the Q buffer.
// ---------------------------------------------------------------------------
__global__ __launch_bounds__(256) void mha_flash_attn(
    const float* __restrict__ Q, const float* __restrict__ Km,
    const float* __restrict__ V, float* __restrict__ Z) {
  __shared__ float ktile[2][16 * KSTR];
  __shared__ float vtile[2][16 * VSTR];
  __shared__ float pbuf[8][16][17];             // per-wave P re-layout tile

  const int tid  = threadIdx.x;
  const int lane = tid & 31;
  const int wave = tid >> 5;
  const int m    = lane & 15;
  const int lh   = lane >> 4;

  const int chunksPerBH = Nn / 128;             // 16
  const int bh    = blockIdx.x / chunksPerBH;
  const int chunk = blockIdx.x % chunksPerBH;
  const int q0    = chunk * 128 + wave * 16;    // first query row for this wave

  const float* Qb = Q  + (size_t)bh * Nn * D;
  const float* Kb = Km + (size_t)bh * Nn * D;
  const float* Vb = V  + (size_t)bh * Nn * D;

  // Cooperative-fill coordinates: 256 threads x 1 float4 = one 16x64 tile
  const int frow = tid >> 4;                    // key row within tile
  const int fc   = (tid & 15) * 4;              // dword column

  // Preload Q A-fragments, pre-scaled by d^-0.5 = 0.125
  v2f qf[16];
#pragma unroll
  for (int s = 0; s < 16; ++s) {
    v2f a = *(const v2f*)(Qb + (size_t)(q0 + m) * D + s * 4 + 2 * lh);
    qf[s].x = a.x * 0.125f;
    qf[s].y = a.y * 0.125f;
  }

  v8f acc[4];
  float mrow[8], lrow[8];
#pragma unroll
  for (int t = 0; t < 4; ++t) acc[t] = (v8f){0.f, 0.f, 0.f, 0.f, 0.f, 0.f, 0.f, 0.f};
#pragma unroll
  for (int i = 0; i < 8; ++i) { mrow[i] = -1.0e30f; lrow[i] = 0.f; }

  // Prologue: stage first K/V tile into buffer 0
#if HAVE_ASYNC_LDS
  async_g2l_b128(Kb + (size_t)frow * D + fc, &ktile[0][frow * KSTR + fc]);
  async_g2l_b128(Vb + (size_t)frow * D + fc, &vtile[0][frow * VSTR + fc]);
  __builtin_amdgcn_s_wait_asynccnt(0);
#else
  {
    v4f kr = *(const v4f*)(Kb + (size_t)frow * D + fc);
    v4f vr = *(const v4f*)(Vb + (size_t)frow * D + fc);
    *(v4f*)(&ktile[0][frow * KSTR + fc]) = kr;
    *(v4f*)(&vtile[0][frow * VSTR + fc]) = vr;
  }
#endif
  __syncthreads();

  const int NB = Nn / 16;                       // 128 key tiles
  for (int jb = 0; jb < NB; ++jb) {
    const int cur = jb & 1;
    const int nxt = cur ^ 1;
    const int j0n = (jb + 1) * 16;

#if HAVE_ASYNC_LDS
    // Barrier ending iteration jb-1 guaranteed all readers of 'nxt' are done:
    // issue the async fill now so it overlaps the whole compute section.
    if (j0n < Nn) {
      async_g2l_b128(Kb + (size_t)(j0n + frow) * D + fc, &ktile[nxt][frow * KSTR + fc]);
      async_g2l_b128(Vb + (size_t)(j0n + frow) * D + fc, &vtile[nxt][frow * VSTR + fc]);
    }
#else
    v4f kr, vr;
    if (j0n < Nn) {
      kr = *(const v4f*)(Kb + (size_t)(j0n + frow) * D + fc);
      vr = *(const v4f*)(Vb + (size_t)(j0n + frow) * D + fc);
    }
#endif

    // S = (Q*scale) @ K^T : 16 wmma steps; K^T B-fragment read from LDS (b64)
    v8f s = {0.f, 0.f, 0.f, 0.f, 0.f, 0.f, 0.f, 0.f};
#pragma unroll
    for (int ss = 0; ss < 16; ++ss) {
      v2f b = *(const v2f*)(&ktile[cur][m * KSTR + ss * 4 + 2 * lh]);
      s = WMMA4F32(qf[ss], b, s);
    }

    // Online softmax: row stats live per lane-half in slot i (row = i + 8*lh)
    float alpha[8];
#pragma unroll
    for (int i = 0; i < 8; ++i) {
      float v = s[i];
      for (int off = 1; off < 16; off <<= 1) v = fmaxf(v, __shfl_xor(v, off, 32));
      const float mn = fmaxf(mrow[i], v);
      alpha[i] = __expf(mrow[i] - mn);
      mrow[i]  = mn;
    }
#pragma unroll
    for (int i = 0; i < 8; ++i) {
      const float p = __expf(s[i] - mrow[i]);
      s[i] = p;
      float t = p;
      for (int off = 1; off < 16; off <<= 1) t += __shfl_xor(t, off, 32);
      lrow[i] = lrow[i] * alpha[i] + t;
    }
#pragma unroll
    for (int t = 0; t < 4; ++t)
#pragma unroll
      for (int i = 0; i < 8; ++i) acc[t][i] *= alpha[i];

    // Re-layout P: C-layout -> LDS -> A-layout (in-order LDS within a wave)
#pragma unroll
    for (int i = 0; i < 8; ++i) pbuf[wave][i + 8 * lh][m] = s[i];

    // acc += P @ V_block : 4 d-tiles x 4 k-steps; V B-fragment read from LDS
#pragma unroll
    for (int t = 0; t < 4; ++t) {
#pragma unroll
      for (int kk = 0; kk < 16; kk += 4) {
        v2f a;
        a.x = pbuf[wave][m][kk + 2 * lh + 0];
        a.y = pbuf[wave][m][kk + 2 * lh + 1];
        v2f b;
        b.x = vtile[cur][(kk + 2 * lh + 0) * VSTR + t * 16 + m];
        b.y = vtile[cur][(kk + 2 * lh + 1) * VSTR + t * 16 + m];
        acc[t] = WMMA4F32(a, b, acc[t]);
      }
    }

#if HAVE_ASYNC_LDS
    __builtin_amdgcn_s_wait_asynccnt(0);        // next tile landed in LDS
#else
    if (j0n < Nn) {
      *(v4f*)(&ktile[nxt][frow * KSTR + fc]) = kr;
      *(v4f*)(&vtile[nxt][frow * VSTR + fc]) = vr;
    }
#endif
    __syncthreads();
  }

  // z = acc / l, written over the Q buffer (head-major layout)
  float inv[8];
#pragma unroll
  for (int i = 0; i < 8; ++i) inv[i] = 1.0f / lrow[i];
  float* Zb = Z + (size_t)bh * Nn * D;
#pragma unroll
  for (int t = 0; t < 4; ++t)
#pragma unroll
    for (int i = 0; i < 8; ++i)
      Zb[(size_t)(q0 + i + 8 * lh) * D + t * 16 + m] = acc[t][i] * inv[i];
}

// ---------------------------------------------------------------------------
// Stage 3: out = z @ W_out + b_out.  z is head-major [b,h,n,d]; gather A
// fragments accordingly. One 16x16 tile per wave, K=512 -> 128 wmma steps.
// ---------------------------------------------------------------------------
__global__ __launch_bounds__(128) void mha_out_proj(
    const float* __restrict__ Zq,    // [b,h,n,d]
    const float* __restrict__ Wout,  // [512, 64]
    const float* __restrict__ bout,  // [64]
    float* __restrict__ out) {       // [B*N, 64]
  const int lane = threadIdx.x & 31;
  const int wave = threadIdx.x >> 5;
  const int m    = lane & 15;
  const int lh   = lane >> 4;

  const int tile = blockIdx.x * 4 + wave;  // (B*N/16) * (64/16) = 4096 tiles
  const int nt   = tile >> 2;
  const int ct   = tile & 3;

  const int r  = nt * 16 + m;              // A-fragment row for this lane
  const int gb = r / Nn;
  const int n  = r - gb * Nn;

  v8f acc = {0.f, 0.f, 0.f, 0.f, 0.f, 0.f, 0.f, 0.f};
#pragma unroll 8
  for (int kk = 0; kk < HD; kk += 4) {
    const int k0 = kk + 2 * lh;
    const int hh = k0 / D;
    const int d  = k0 - hh * D;            // d, d+1 stay inside one head
    v2f a = *(const v2f*)(Zq + (size_t)((gb * H + hh) * Nn + n) * D + d);
    v2f b;
    b.x = Wout[(size_t)(k0 + 0) * D + ct * 16 + m];
    b.y = Wout[(size_t)(k0 + 1) * D + ct * 16 + m];
    acc = WMMA4F32(a, b, acc);
  }

  const float bias = bout[ct * 16 + m];
  const int rbase  = nt * 16;
#pragma unroll
  for (int i = 0; i < 8; ++i)
    out[(size_t)(rbase + i + 8 * lh) * D + ct * 16 + m] = acc[i] + bias;
}

// ---------------------------------------------------------------------------
extern "C" void kernel_launch(void* const* d_in, const int* in_sizes, int n_in,
                              void* d_out, int out_size, void* d_ws, size_t ws_size,
                              hipStream_t stream) {
  const float* x    = (const float*)d_in[0];
  const float* Wqkv = (const float*)d_in[1];
  const float* Wout = (const float*)d_in[2];
  const float* bout = (const float*)d_in[3];
  float* out = (float*)d_out;

  const size_t BNHD = (size_t)Bc * Nn * H * D;  // 8.39M floats per buffer
  float* Qbuf = (float*)d_ws;
  float* Kbuf = Qbuf + BNHD;
  float* Vbuf = Kbuf + BNHD;

  // Stage 1: 1024 row-tiles * 96 col-tiles, 4 waves/WG
  {
    const int tiles = (Bc * Nn / 16) * (QKV / 16);
    mha_qkv_proj<<<dim3(tiles / 4), dim3(128), 0, stream>>>(x, Wqkv, Qbuf, Kbuf, Vbuf);
  }
  // Stage 2: (B*H) * (N/128) workgroups, 8 waves of 16 query rows each
  {
    const int blocks = (Bc * H) * (Nn / 128);
    mha_flash_attn<<<dim3(blocks), dim3(256), 0, stream>>>(Qbuf, Kbuf, Vbuf, Qbuf);
  }
  // Stage 3: 1024 row-tiles * 4 col-tiles, 4 waves/WG
  {
    const int tiles = (Bc * Nn / 16) * (D / 16);
    mha_out_proj<<<dim3(tiles / 4), dim3(128), 0, stream>>>(Qbuf, Wout, bout, out);
  }
}